// DecoderWithAttention_5162550690713
// MI455X (gfx1250) — compile-verified
//
#include <hip/hip_runtime.h>
#include <hip/hip_bf16.h>
#include <math.h>

// ---------------------------------------------------------------------------
// Sizes (static per the reference)
// ---------------------------------------------------------------------------
#define BATCH   32
#define ENC_C   512
#define P_POS   196          // 14*14
#define ATT_D   512
#define EMB_D   512
#define DEC_D   512
#define VOCAB_N 10000
#define MAXLEN  64
#define TSTEPS  63           // MAXLEN - 1

typedef float v2f __attribute__((ext_vector_type(2)));
typedef float v8f __attribute__((ext_vector_type(8)));

__device__ __forceinline__ float sigf(float v) { return 1.f / (1.f + expf(-v)); }

// ---------------------------------------------------------------------------
// WMMA core: one wave computes a 16x64 fp32 tile via V_WMMA_F32_16X16X4_F32.
// A row-major [M,K], B row-major [K,LDB].
// A 16x4 fragment: lane&15 = M row; lane<16 holds K={0,1}, lane>=16 K={2,3}.
// B 4x16 fragment: lane&15 = N col; same K split.
// C/D: vgpr r -> row r + 8*(lane>=16), col = lane&15.
// Straight-line, branch-free: EXEC stays all-ones through every WMMA.
// ---------------------------------------------------------------------------
template<int K, int LDB>
__device__ __forceinline__ void wmma_k_loop(const float* __restrict__ A_row,
                                            const float* __restrict__ B,
                                            const int nc[4], int khalf,
                                            v8f acc[4])
{
#pragma unroll 2
    for (int k = 0; k < K; k += 4) {
        const v2f a = *(const v2f*)(A_row + k + khalf);   // 8B-aligned pair
#pragma unroll
        for (int j = 0; j < 4; ++j) {
            v2f bb;
            bb.x = B[(size_t)(k + khalf)     * LDB + nc[j]];
            bb.y = B[(size_t)(k + khalf + 1) * LDB + nc[j]];
            acc[j] = __builtin_amdgcn_wmma_f32_16x16x4_f32(
                false, a, false, bb, (short)0, acc[j], false, false);
        }
    }
}

// Generic templated GEMM: C[M,N] = A[M,K] @ B[K,N] + bias[N]
// One wave per 16x64 tile; N tail handled by clamping (redundant identical
// stores, no divergence). Requires M%16==0, N%16==0, K%4==0.
template<int M, int N, int K>
__global__ void gemm_wmma_kernel(const float* __restrict__ A,
                                 const float* __restrict__ B,
                                 const float* __restrict__ bias,
                                 float* __restrict__ C)
{
    constexpr int NGROUPS = (N + 63) / 64;
    constexpr int MTILES  = M / 16;
    const int wave = __builtin_amdgcn_readfirstlane(
        (int)((blockIdx.x * blockDim.x + threadIdx.x) >> 5));
    if (wave >= MTILES * NGROUPS) return;           // scalar branch
    const int lane  = (int)(threadIdx.x & 31);
    const int m0    = (wave / NGROUPS) * 16;
    const int n0    = (wave % NGROUPS) * 64;
    const int col   = lane & 15;
    const int khalf = (lane >> 4) * 2;
    const int hi8   = (lane >> 4) * 8;

    int ncb[4], nc[4];
#pragma unroll
    for (int j = 0; j < 4; ++j) {
        int b0 = n0 + j * 16;
        if (b0 > N - 16) b0 = N - 16;               // clamp tail (uniform value)
        ncb[j] = b0;
        nc[j]  = b0 + col;
    }

    v8f zero = {0.f,0.f,0.f,0.f,0.f,0.f,0.f,0.f};
    v8f acc[4] = {zero, zero, zero, zero};

    wmma_k_loop<K, N>(A + (size_t)(m0 + (lane & 15)) * K, B, nc, khalf, acc);

#pragma unroll
    for (int j = 0; j < 4; ++j) {
        const int c = ncb[j] + col;
        const float bv = bias[c];
#pragma unroll
        for (int r = 0; r < 8; ++r)
            C[(size_t)(m0 + r + hi8) * N + c] = acc[j][r] + bv;
    }
}

// LSTM gates: gates[32,2048] = x[32,1024]@W_ih + h[32,512]@W_hh + b_ih + b_hh
__global__ void gates_wmma_kernel(const float* __restrict__ x,
                                  const float* __restrict__ W_ih,
                                  const float* __restrict__ h,
                                  const float* __restrict__ W_hh,
                                  const float* __restrict__ b_ih,
                                  const float* __restrict__ b_hh,
                                  float* __restrict__ gates)
{
    constexpr int N = 4 * DEC_D;       // 2048
    constexpr int NG = N / 64;         // 32 column groups, 64 waves total
    const int wave = __builtin_amdgcn_readfirstlane(
        (int)((blockIdx.x * blockDim.x + threadIdx.x) >> 5));
    if (wave >= 2 * NG) return;
    const int lane  = (int)(threadIdx.x & 31);
    const int m0    = (wave / NG) * 16;
    const int n0    = (wave % NG) * 64;
    const int col   = lane & 15;
    const int khalf = (lane >> 4) * 2;
    const int hi8   = (lane >> 4) * 8;

    int nc[4];
#pragma unroll
    for (int j = 0; j < 4; ++j) nc[j] = n0 + j * 16 + col;

    v8f zero = {0.f,0.f,0.f,0.f,0.f,0.f,0.f,0.f};
    v8f acc[4] = {zero, zero, zero, zero};

    wmma_k_loop<EMB_D + ENC_C, N>(x + (size_t)(m0 + col) * (EMB_D + ENC_C), W_ih, nc, khalf, acc);
    wmma_k_loop<DEC_D,         N>(h + (size_t)(m0 + col) * DEC_D,           W_hh, nc, khalf, acc);

#pragma unroll
    for (int j = 0; j < 4; ++j) {
        const int c = n0 + j * 16 + col;
        const float bv = b_ih[c] + b_hh[c];
#pragma unroll
        for (int r = 0; r < 8; ++r)
            gates[(size_t)(m0 + r + hi8) * N + c] = acc[j][r] + bv;
    }
}

// Batched FC over all saved hidden states:
// out[b,t,:] = (t < len[b]-1) ? hsave[t*32+b,:] @ W_fc + b_fc : 0
__global__ void fc_wmma_kernel(const float* __restrict__ hsave,   // [2016,512]
                               const float* __restrict__ W_fc,    // [512,10000]
                               const float* __restrict__ b_fc,    // [10000]
                               const int*   __restrict__ cap_len, // [32]
                               float* __restrict__ out)           // [32,63,10000]
{
    constexpr int N = VOCAB_N, K = DEC_D;
    constexpr int NGROUPS = (N + 63) / 64;             // 157
    constexpr int MTILES  = (TSTEPS * BATCH) / 16;     // 126
    const int wave = __builtin_amdgcn_readfirstlane(
        (int)((blockIdx.x * blockDim.x + threadIdx.x) >> 5));
    if (wave >= MTILES * NGROUPS) return;
    const int lane  = (int)(threadIdx.x & 31);
    const int m0    = (wave / NGROUPS) * 16;
    const int n0    = (wave % NGROUPS) * 64;
    const int col   = lane & 15;
    const int khalf = (lane >> 4) * 2;
    const int hi8   = (lane >> 4) * 8;

    int ncb[4], nc[4];
#pragma unroll
    for (int j = 0; j < 4; ++j) {
        int b0 = n0 + j * 16;
        if (b0 > N - 16) b0 = N - 16;                  // clamp tail
        ncb[j] = b0;
        nc[j]  = b0 + col;
    }

    v8f zero = {0.f,0.f,0.f,0.f,0.f,0.f,0.f,0.f};
    v8f acc[4] = {zero, zero, zero, zero};

    wmma_k_loop<K, N>(hsave + (size_t)(m0 + col) * K, W_fc, nc, khalf, acc);

#pragma unroll
    for (int j = 0; j < 4; ++j) {
        const int c = ncb[j] + col;
        const float bv = b_fc[c];
#pragma unroll
        for (int r = 0; r < 8; ++r) {
            const int row = m0 + r + hi8;              // row = t*32 + b
            const int t = row >> 5;
            const int b = row & 31;
            const float v = (t < (cap_len[b] - 1)) ? (acc[j][r] + bv) : 0.f;
            out[((size_t)b * TSTEPS + t) * VOCAB_N + c] = v;
        }
    }
}

// ---------------------------------------------------------------------------
// Non-matrix kernels
// ---------------------------------------------------------------------------

// enc[b,p,c] = encoder_out[b,c,p]; mean[b,c] = mean_p enc[b,p,c]
__global__ void prep_kernel(const float* __restrict__ eo,   // [32,512,196]
                            float* __restrict__ enc,        // [32,196,512]
                            float* __restrict__ mean)       // [32,512]
{
    const int b = blockIdx.x, tid = threadIdx.x;            // 256 threads
    for (int c = tid; c < ENC_C; c += 256) {
        const float* src = eo + ((size_t)b * ENC_C + c) * P_POS;
        float s = 0.f;
        for (int p = 0; p < P_POS; ++p) s += src[p];
        mean[(size_t)b * ENC_C + c] = s * (1.f / (float)P_POS);
    }
    for (int idx = tid; idx < P_POS * ENC_C; idx += 256) {
        const int p = idx / ENC_C, c = idx % ENC_C;
        enc[((size_t)b * P_POS + p) * ENC_C + c] =
            eo[((size_t)b * ENC_C + c) * P_POS + p];
    }
}

// Per-batch attention: score -> softmax -> awe -> gated x
__global__ void att_step_kernel(const float* __restrict__ enc,      // [32,196,512]
                                const float* __restrict__ enc_att,  // [32,196,512]
                                const float* __restrict__ dec_a,    // [32,512]
                                const float* __restrict__ gatepre,  // [32,512]
                                const float* __restrict__ w_full,   // [512]
                                const float* __restrict__ b_full,   // [1]
                                const float* __restrict__ emb_table,// [10000,512]
                                const int*   __restrict__ captions, // [32,64]
                                float* __restrict__ x,              // [32,1024]
                                int t)
{
    const int b = blockIdx.x, tid = threadIdx.x;            // 256 threads
    __shared__ float sda[ATT_D];
    __shared__ float swf[ATT_D];
    __shared__ float ssc[256];
    __shared__ float red[256];

    for (int i = tid; i < ATT_D; i += 256) {
        sda[i] = dec_a[(size_t)b * ATT_D + i];
        swf[i] = w_full[i];
    }
    __syncthreads();

    float score = -INFINITY;
    if (tid < P_POS) {
        const float* ea = enc_att + ((size_t)b * P_POS + tid) * ATT_D;
        float s = 0.f;
        for (int a = 0; a < ATT_D; ++a) {
            const float v = ea[a] + sda[a];
            s += (v > 0.f ? v : 0.f) * swf[a];
        }
        score = s + b_full[0];
    }
    ssc[tid] = score;
    red[tid] = score;
    __syncthreads();
    for (int off = 128; off > 0; off >>= 1) {
        if (tid < off) red[tid] = fmaxf(red[tid], red[tid + off]);
        __syncthreads();
    }
    const float mx = red[0];
    __syncthreads();
    const float e = (tid < P_POS) ? expf(ssc[tid] - mx) : 0.f;
    red[tid] = e;
    __syncthreads();
    for (int off = 128; off > 0; off >>= 1) {
        if (tid < off) red[tid] += red[tid + off];
        __syncthreads();
    }
    const float inv = 1.f / red[0];
    ssc[tid] = e * inv;                // alpha (0 for tid >= 196)
    __syncthreads();

    const int cap = captions[b * MAXLEN + t];
    const float* eb = emb_table + (size_t)cap * EMB_D;
    for (int c = tid; c < ENC_C; c += 256) {
        float aw = 0.f;
        const float* ep = enc + (size_t)b * P_POS * ENC_C + c;
        for (int p = 0; p < P_POS; ++p) aw += ssc[p] * ep[(size_t)p * ENC_C];
        const float g = sigf(gatepre[(size_t)b * ENC_C + c]);
        x[(size_t)b * (EMB_D + ENC_C) + EMB_D + c] = g * aw;
        x[(size_t)b * (EMB_D + ENC_C) + c]         = eb[c];
    }
}

// LSTM pointwise update + masked state carry + save h for batched FC
__global__ void lstm_step_kernel(const float* __restrict__ gates,   // [32,2048]
                                 float* __restrict__ h,             // [32,512]
                                 float* __restrict__ c,             // [32,512]
                                 float* __restrict__ hsave,         // [63*32,512]
                                 const int* __restrict__ cap_len,   // [32]
                                 int t)
{
    const int b = blockIdx.x, d = threadIdx.x;              // 512 threads
    const float* g = gates + (size_t)b * (4 * DEC_D);
    const float iv = sigf(g[d]);
    const float fv = sigf(g[DEC_D + d]);
    const float gv = tanhf(g[2 * DEC_D + d]);
    const float ov = sigf(g[3 * DEC_D + d]);
    const float cn = fv * c[(size_t)b * DEC_D + d] + iv * gv;
    const float hn = ov * tanhf(cn);
    const bool mask = t < (cap_len[b] - 1);
    if (mask) {
        h[(size_t)b * DEC_D + d] = hn;
        c[(size_t)b * DEC_D + d] = cn;
    }
    hsave[((size_t)t * BATCH + b) * DEC_D + d] = hn;
}

// ---------------------------------------------------------------------------
// Host launcher
// ---------------------------------------------------------------------------
static inline int gemm_blocks_rt(int M, int N) {
    const int tiles = (M / 16) * ((N + 63) / 64);
    return (tiles + 7) / 8;   // 8 waves per 256-thread block
}

extern "C" void kernel_launch(void* const* d_in, const int* in_sizes, int n_in,
                              void* d_out, int out_size, void* d_ws, size_t ws_size,
                              hipStream_t stream) {
    (void)in_sizes; (void)n_in; (void)out_size; (void)ws_size;

    const float* encoder_out = (const float*)d_in[0];
    const int*   captions    = (const int*)  d_in[1];
    const int*   cap_len     = (const int*)  d_in[2];
    const float* emb_table   = (const float*)d_in[3];
    const float* W_enc_att   = (const float*)d_in[4];
    const float* b_enc_att   = (const float*)d_in[5];
    const float* W_dec_att   = (const float*)d_in[6];
    const float* b_dec_att   = (const float*)d_in[7];
    const float* W_full_att  = (const float*)d_in[8];
    const float* b_full_att  = (const float*)d_in[9];
    const float* W_init_h    = (const float*)d_in[10];
    const float* b_init_h    = (const float*)d_in[11];
    const float* W_init_c    = (const float*)d_in[12];
    const float* b_init_c    = (const float*)d_in[13];
    const float* W_f_beta    = (const float*)d_in[14];
    const float* b_f_beta    = (const float*)d_in[15];
    const float* W_ih        = (const float*)d_in[16];
    const float* b_ih        = (const float*)d_in[17];
    const float* W_hh        = (const float*)d_in[18];
    const float* b_hh        = (const float*)d_in[19];
    const float* W_fc        = (const float*)d_in[20];
    const float* b_fc        = (const float*)d_in[21];
    float* out = (float*)d_out;

    // Workspace layout (floats) — ~30.5 MB total
    float* ws      = (float*)d_ws;
    float* enc     = ws;                                  // 32*196*512
    float* enc_att = enc     + (size_t)BATCH * P_POS * ENC_C;
    float* mean    = enc_att + (size_t)BATCH * P_POS * ATT_D;  // 32*512
    float* h       = mean    + (size_t)BATCH * ENC_C;
    float* c       = h       + (size_t)BATCH * DEC_D;
    float* dec_a   = c       + (size_t)BATCH * DEC_D;
    float* gatepre = dec_a   + (size_t)BATCH * ATT_D;
    float* xbuf    = gatepre + (size_t)BATCH * ENC_C;     // 32*1024
    float* gates   = xbuf    + (size_t)BATCH * (EMB_D + ENC_C); // 32*2048
    float* hsave   = gates   + (size_t)BATCH * 4 * DEC_D; // 63*32*512

    // ---- one-time prep ----
    prep_kernel<<<BATCH, 256, 0, stream>>>(encoder_out, enc, mean);

    gemm_wmma_kernel<BATCH, DEC_D, ENC_C><<<gemm_blocks_rt(BATCH, DEC_D), 256, 0, stream>>>(
        mean, W_init_h, b_init_h, h);
    gemm_wmma_kernel<BATCH, DEC_D, ENC_C><<<gemm_blocks_rt(BATCH, DEC_D), 256, 0, stream>>>(
        mean, W_init_c, b_init_c, c);
    gemm_wmma_kernel<BATCH * P_POS, ATT_D, ENC_C><<<gemm_blocks_rt(BATCH * P_POS, ATT_D), 256, 0, stream>>>(
        enc, W_enc_att, b_enc_att, enc_att);

    // ---- 63-step recurrence ----
    for (int t = 0; t < TSTEPS; ++t) {
        gemm_wmma_kernel<BATCH, ATT_D, DEC_D><<<gemm_blocks_rt(BATCH, ATT_D), 256, 0, stream>>>(
            h, W_dec_att, b_dec_att, dec_a);
        gemm_wmma_kernel<BATCH, ENC_C, DEC_D><<<gemm_blocks_rt(BATCH, ENC_C), 256, 0, stream>>>(
            h, W_f_beta, b_f_beta, gatepre);
        att_step_kernel<<<BATCH, 256, 0, stream>>>(
            enc, enc_att, dec_a, gatepre, W_full_att, b_full_att,
            emb_table, captions, xbuf, t);
        gates_wmma_kernel<<<8, 256, 0, stream>>>(
            xbuf, W_ih, h, W_hh, b_ih, b_hh, gates);
        lstm_step_kernel<<<BATCH, DEC_D, 0, stream>>>(
            gates, h, c, hsave, cap_len, t);
    }

    // ---- batched FC over all timesteps (hoisted off the serial chain) ----
    fc_wmma_kernel<<<gemm_blocks_rt(TSTEPS * BATCH, VOCAB_N), 256, 0, stream>>>(
        hsave, W_fc, b_fc, cap_len, out);
}